// GC2N_37160057045292
// MI455X (gfx1250) — compile-verified
//
#include <hip/hip_runtime.h>
#include <math.h>

typedef __attribute__((ext_vector_type(2))) float v2f;
typedef __attribute__((ext_vector_type(8))) float v8f;

// ---------------------------------------------------------------------------
// WMMA f32 GEMM: C[M,N] = A[M,K] * B[K,N]
// One wave computes a 16x64 output strip: 4 accumulators, so each A fragment
// feeds 4 independent V_WMMA_F32_16X16X4_F32 chains. Requires M%16==0,
// K%4==0, N%64==0 (true here: M=50000=16*3125, K in {128,256}, N in {256,128}).
// A-fragment layout (ISA 7.12.2, 32-bit A 16x4): lanes 0-15 hold K=kk..kk+1,
// lanes 16-31 hold K=kk+2..kk+3, row M = lane%16.
// C/D layout: VGPR v -> row (v + 8*(lane/16)), col = lane%16.
// ---------------------------------------------------------------------------
__global__ void gcn_gemm_wmma_f32(const float* __restrict__ A,
                                  const float* __restrict__ B,
                                  float* __restrict__ C,
                                  int M, int K, int N) {
    const int wavesPerBlock = blockDim.x >> 5;
    const int wave = threadIdx.x >> 5;
    const int lane = threadIdx.x & 31;
    const int half = lane >> 4;       // 0: lanes 0-15, 1: lanes 16-31
    const int lm   = lane & 15;       // row/col within tile

    const int tileM  = blockIdx.x;
    const int tileN4 = blockIdx.y * wavesPerBlock + wave;   // 64-wide N chunk
    if (tileM * 16 >= M || tileN4 * 64 >= N) return;        // wave-uniform guard

    const float* Arow = A + (size_t)(tileM * 16 + lm) * K;  // this lane's A row
    const float* B0   = B + (size_t)tileN4 * 64 + lm;       // this lane's B cols

    v8f acc0 = {}, acc1 = {}, acc2 = {}, acc3 = {};
    for (int kk = 0; kk < K; kk += 4) {
        const int ka = kk + 2 * half;
        v2f a;
        a.x = Arow[ka];
        a.y = Arow[ka + 1];
        const float* Bk0 = B0 + (size_t)ka * N;
        const float* Bk1 = B0 + (size_t)(ka + 1) * N;
        v2f b0, b1, b2, b3;
        b0.x = Bk0[0];  b0.y = Bk1[0];
        b1.x = Bk0[16]; b1.y = Bk1[16];
        b2.x = Bk0[32]; b2.y = Bk1[32];
        b3.x = Bk0[48]; b3.y = Bk1[48];
        acc0 = __builtin_amdgcn_wmma_f32_16x16x4_f32(false, a, false, b0, (short)0, acc0, false, false);
        acc1 = __builtin_amdgcn_wmma_f32_16x16x4_f32(false, a, false, b1, (short)0, acc1, false, false);
        acc2 = __builtin_amdgcn_wmma_f32_16x16x4_f32(false, a, false, b2, (short)0, acc2, false, false);
        acc3 = __builtin_amdgcn_wmma_f32_16x16x4_f32(false, a, false, b3, (short)0, acc3, false, false);
    }

    float* Crow = C + (size_t)(tileM * 16 + 8 * half) * N + (size_t)tileN4 * 64 + lm;
#pragma unroll
    for (int v = 0; v < 8; ++v) {
        float* cr = Crow + (size_t)v * N;
        cr[0]  = acc0[v];
        cr[16] = acc1[v];
        cr[32] = acc2[v];
        cr[48] = acc3[v];
    }
}

// ---------------------------------------------------------------------------
// Degree / normalization kernels
// ---------------------------------------------------------------------------
__global__ void gcn_zero_f32(float* __restrict__ p, int n) {
    int i = blockIdx.x * blockDim.x + threadIdx.x;
    if (i < n) p[i] = 0.0f;
}

__global__ void gcn_deg_accum(const long long* __restrict__ col,
                              const float* __restrict__ ew,
                              float* __restrict__ deg, int E) {
    int e = blockIdx.x * blockDim.x + threadIdx.x;
    if (e < E) atomicAdd(&deg[(int)col[e]], ew[e]);
}

__global__ void gcn_dinv(float* __restrict__ deg, int n) {
    int i = blockIdx.x * blockDim.x + threadIdx.x;
    if (i < n) {
        float d = deg[i] + 1.0f;              // + self-loop weight
        deg[i] = (d > 0.0f) ? rsqrtf(d) : 0.0f;
    }
}

__global__ void gcn_edge_norm(const long long* __restrict__ row,
                              const long long* __restrict__ col,
                              const float* __restrict__ ew,
                              const float* __restrict__ dinv,
                              float* __restrict__ norm, int E) {
    int e = blockIdx.x * blockDim.x + threadIdx.x;
    if (e < E) norm[e] = dinv[(int)row[e]] * ew[e] * dinv[(int)col[e]];
}

// agg[i][:] = dinv[i]^2 * h[i][:]   (self-loop contribution doubles as init)
__global__ void gcn_selfloop_init(const float4* __restrict__ h,
                                  const float* __restrict__ dinv,
                                  float4* __restrict__ agg,
                                  int n, int Fv) {
    int idx = blockIdx.x * blockDim.x + threadIdx.x;
    if (idx < n * Fv) {
        int i = idx / Fv;
        float s = dinv[i] * dinv[i];
        float4 v = h[idx];
        float4 r; r.x = v.x * s; r.y = v.y * s; r.z = v.z * s; r.w = v.w * s;
        agg[idx] = r;
    }
}

// One wave per edge: agg[col][:] += norm[e] * h[row][:]
__global__ void gcn_agg_edges(const long long* __restrict__ row,
                              const long long* __restrict__ col,
                              const float* __restrict__ norm,
                              const float* __restrict__ h,
                              float* __restrict__ agg,
                              int E, int F) {
    int wave = (int)((blockIdx.x * (size_t)blockDim.x + threadIdx.x) >> 5);
    int lane = threadIdx.x & 31;
    if (wave >= E) return;
    int r = (int)row[wave];
    int c = (int)col[wave];
    float nm = norm[wave];
    const float* hs = h + (size_t)r * F;
    float* ad = agg + (size_t)c * F;
    for (int f = lane * 4; f < F; f += 128) {
        float4 v = *(const float4*)(hs + f);
        atomicAdd(ad + f + 0, nm * v.x);
        atomicAdd(ad + f + 1, nm * v.y);
        atomicAdd(ad + f + 2, nm * v.z);
        atomicAdd(ad + f + 3, nm * v.w);
    }
}

// out[i][:] = relu(agg[i][:] + bias[:])
__global__ void gcn_bias_relu(const float4* __restrict__ agg,
                              const float* __restrict__ bias,
                              float4* __restrict__ out,
                              int n, int Fv) {
    int idx = blockIdx.x * blockDim.x + threadIdx.x;
    if (idx < n * Fv) {
        float4 v = agg[idx];
        float4 b = ((const float4*)bias)[idx % Fv];
        float4 r;
        r.x = fmaxf(v.x + b.x, 0.0f);
        r.y = fmaxf(v.y + b.y, 0.0f);
        r.z = fmaxf(v.z + b.z, 0.0f);
        r.w = fmaxf(v.w + b.w, 0.0f);
        out[idx] = r;
    }
}

// ---------------------------------------------------------------------------
extern "C" void kernel_launch(void* const* d_in, const int* in_sizes, int n_in,
                              void* d_out, int out_size, void* d_ws, size_t ws_size,
                              hipStream_t stream) {
    const float*     x   = (const float*)d_in[0];
    const long long* ei  = (const long long*)d_in[1];
    const float*     ew  = (const float*)d_in[2];
    const float*     W1  = (const float*)d_in[3];
    const float*     b1  = (const float*)d_in[4];
    const float*     W2  = (const float*)d_in[5];
    const float*     b2  = (const float*)d_in[6];
    float*           out = (float*)d_out;

    const int H = in_sizes[4];                 // hidden dim (256)
    const int I = in_sizes[3] / H;             // in dim (128)
    const int O = in_sizes[6];                 // out dim (128)
    const int N = in_sizes[0] / I;             // nodes (50000)
    const int E = in_sizes[2];                 // edges (600000)

    const long long* row = ei;
    const long long* col = ei + E;

    // Workspace layout (floats). All section sizes are 16B-aligned here.
    float* ws    = (float*)d_ws;
    float* dinv  = ws;                          // N   (deg, then dinv in place)
    float* norm  = dinv + N;                    // E
    float* hbuf  = norm + E;                    // N*H
    float* abuf  = hbuf + (size_t)N * H;        // N*H

    const int T = 256;
    // ---- normalization coefficients ----
    gcn_zero_f32 <<<(N + T - 1) / T, T, 0, stream>>>(dinv, N);
    gcn_deg_accum<<<(E + T - 1) / T, T, 0, stream>>>(col, ew, dinv, E);
    gcn_dinv     <<<(N + T - 1) / T, T, 0, stream>>>(dinv, N);
    gcn_edge_norm<<<(E + T - 1) / T, T, 0, stream>>>(row, col, ew, dinv, norm, E);

    // ---- layer 1: h1 = x @ W1 ----
    {
        dim3 block(128);                                    // 4 waves = 4 N-chunks
        dim3 grid((N + 15) / 16, (H / 64 + 3) / 4);
        gcn_gemm_wmma_f32<<<grid, block, 0, stream>>>(x, W1, hbuf, N, I, H);
    }
    {
        int total = N * (H / 4);
        gcn_selfloop_init<<<(total + T - 1) / T, T, 0, stream>>>(
            (const float4*)hbuf, dinv, (float4*)abuf, N, H / 4);
    }
    gcn_agg_edges<<<(E + 7) / 8, T, 0, stream>>>(row, col, norm, hbuf, abuf, E, H);
    {
        int total = N * (H / 4);
        gcn_bias_relu<<<(total + T - 1) / T, T, 0, stream>>>(
            (const float4*)abuf, b1, (float4*)hbuf, N, H / 4);   // h1act -> hbuf
    }

    // ---- layer 2: h2 = h1act @ W2 ----
    {
        dim3 block(128);
        dim3 grid((N + 15) / 16, (O / 64 + 3) / 4);
        gcn_gemm_wmma_f32<<<grid, block, 0, stream>>>(hbuf, W2, abuf, N, H, O); // h2 -> abuf[N*O]
    }
    {
        int total = N * (O / 4);
        gcn_selfloop_init<<<(total + T - 1) / T, T, 0, stream>>>(
            (const float4*)abuf, dinv, (float4*)hbuf, N, O / 4); // agg2 -> hbuf[N*O]
    }
    gcn_agg_edges<<<(E + 7) / 8, T, 0, stream>>>(row, col, norm, abuf, hbuf, E, O);
    {
        int total = N * (O / 4);
        gcn_bias_relu<<<(total + T - 1) / T, T, 0, stream>>>(
            (const float4*)hbuf, b2, (float4*)out, N, O / 4);
    }
}